// GNN_7_49520972923184
// MI455X (gfx1250) — compile-verified
//
#include <hip/hip_runtime.h>

typedef __attribute__((ext_vector_type(2))) float v2f;
typedef __attribute__((ext_vector_type(8))) float v8f;

#define THREADS 256

// ---------------------------------------------------------------- utilities
__global__ void zero_f32(float* __restrict__ p, int n) {
    int i = blockIdx.x * blockDim.x + threadIdx.x;
    if (i < n) p[i] = 0.0f;
}

// -------------------------------------------------- edge gather/scatter-add
// agg[dst[e], c] += h[src[e], c] * w[e]   (fp32 atomics -> L2-resident, 41MB max)
__global__ void edge_scatter(const float* __restrict__ h,
                             const int*   __restrict__ src,
                             const int*   __restrict__ dst,
                             const float* __restrict__ w,
                             float*       __restrict__ agg,
                             int E, int C) {
    int idx = blockIdx.x * blockDim.x + threadIdx.x;
    if (idx >= E * C) return;
    int e = idx / C;
    int c = idx - e * C;
    float v = h[src[e] * C + c] * w[e];
    atomicAdd(&agg[dst[e] * C + c], v);
}

// ------------------------------------------------------------- GCN GEMM+ReLU
// hout = relu(agg @ Wrel + hin @ Wroot + b)
// One wave32 computes one 16x16 output tile via V_WMMA_F32_16X16X4_F32.
// A layout (16x4 f32): lane<16 -> row=lane, K={k,k+1}; lane>=16 -> row=lane-16, K={k+2,k+3}
// B layout (4x16 f32): mirrored over N.  C/D: vgpr j -> row j+8*(lane>=16), col lane&15.
__global__ void gconv_gemm(const float* __restrict__ agg,
                           const float* __restrict__ hin,
                           const float* __restrict__ Wrel,
                           const float* __restrict__ Wroot,
                           const float* __restrict__ bias,
                           float*       __restrict__ hout,
                           int M, int fin, int fout) {
    const int lane   = threadIdx.x & 31;
    const int wave   = threadIdx.x >> 5;
    const int tilesN = fout >> 4;
    const int tile   = blockIdx.x * (blockDim.x >> 5) + wave;
    const int total  = (M >> 4) * tilesN;
    if (tile >= total) return;           // uniform per wave: EXEC all-1 below

    const int tm = tile / tilesN;
    const int tn = tile - tm * tilesN;
    const int m0 = tm << 4;
    const int n0 = tn << 4;

    const int rl = lane & 15;            // row (A) / col (B)
    const int hi = lane >> 4;            // K-half select

    v8f acc = {};
    const float* arow0 = agg + (size_t)(m0 + rl) * fin + 2 * hi;
    const float* arow1 = hin + (size_t)(m0 + rl) * fin + 2 * hi;

    for (int k = 0; k < fin; k += 4) {
        v2f a, b;
        // lin_rel contribution: A = agg tile, B = Wrel tile
        a.x = arow0[k];
        a.y = arow0[k + 1];
        const float* bp = Wrel + (size_t)(k + 2 * hi) * fout + n0 + rl;
        b.x = bp[0];
        b.y = bp[fout];
        acc = __builtin_amdgcn_wmma_f32_16x16x4_f32(false, a, false, b,
                                                    (short)0, acc, false, false);
        // lin_root contribution: A = hin tile, B = Wroot tile
        a.x = arow1[k];
        a.y = arow1[k + 1];
        const float* bq = Wroot + (size_t)(k + 2 * hi) * fout + n0 + rl;
        b.x = bq[0];
        b.y = bq[fout];
        acc = __builtin_amdgcn_wmma_f32_16x16x4_f32(false, a, false, b,
                                                    (short)0, acc, false, false);
    }

    const float bv = bias[n0 + rl];
#pragma unroll
    for (int j = 0; j < 8; ++j) {
        int row = m0 + j + 8 * hi;
        float v = acc[j] + bv;
        v = fmaxf(v, 0.0f);
        hout[(size_t)row * fout + n0 + rl] = v;
    }
}

// ------------------------------------------------------------- mean pooling
__global__ void pool_sum(const float* __restrict__ h,
                         const int*   __restrict__ batch,
                         float* __restrict__ gsum,
                         float* __restrict__ counts,
                         int N, int C) {
    int idx = blockIdx.x * blockDim.x + threadIdx.x;
    if (idx >= N * C) return;
    int n = idx / C;
    int c = idx - n * C;
    int b = batch[n];
    atomicAdd(&gsum[b * C + c], h[n * C + c]);
    if (c == 0) atomicAdd(&counts[b], 1.0f);
}

__global__ void pool_div(float* __restrict__ gsum,
                         const float* __restrict__ counts,
                         int G, int C) {
    int idx = blockIdx.x * blockDim.x + threadIdx.x;
    if (idx >= G * C) return;
    int g = idx / C;
    gsum[idx] = gsum[idx] / fmaxf(counts[g], 1.0f);
}

// ---------------------------------------------------------------- tiny MLP
__global__ void mlp_layer(const float* __restrict__ in,
                          const float* __restrict__ W,
                          const float* __restrict__ b,
                          float* __restrict__ out,
                          int G, int fin, int fout, int relu) {
    int idx = blockIdx.x * blockDim.x + threadIdx.x;
    if (idx >= G * fout) return;
    int g = idx / fout;
    int o = idx - g * fout;
    float s = b[o];
    const float* xi = in + (size_t)g * fin;
    for (int k = 0; k < fin; ++k) s = fmaf(xi[k], W[(size_t)k * fout + o], s);
    if (relu) s = fmaxf(s, 0.0f);
    out[idx] = s;
}

static inline int cdiv(long long a, int b) { return (int)((a + b - 1) / b); }

extern "C" void kernel_launch(void* const* d_in, const int* in_sizes, int n_in,
                              void* d_out, int out_size, void* d_ws, size_t ws_size,
                              hipStream_t stream) {
    const float* x   = (const float*)d_in[0];
    const int*   ei  = (const int*)d_in[1];
    const float* ea  = (const float*)d_in[2];
    const int*   bat = (const int*)d_in[3];

    const int N = in_sizes[3];       // 20000 nodes
    const int E = in_sizes[2];       // 160000 edges
    const int G = 128;               // graphs
    const int src_off = 0, dst_off = E;

    static const int DIMS[8] = {4, 32, 128, 256, 512, 512, 256, 256};

    // workspace layout (floats)
    float* ws   = (float*)d_ws;
    const size_t HB = (size_t)N * 512;   // max node-feature buffer
    float* hA     = ws;
    float* hB     = ws + HB;
    float* agg    = ws + 2 * HB;
    float* gbuf0  = ws + 3 * HB;                 // 128*256
    float* gbuf1  = gbuf0 + (size_t)G * 256;     // 128*256
    float* counts = gbuf1 + (size_t)G * 256;     // 128
    (void)ws_size; (void)n_in; (void)out_size;

    const float* hin = x;
    float* hout = hA;

    for (int l = 0; l < 7; ++l) {
        const int fin  = DIMS[l];
        const int fout = DIMS[l + 1];
        const float* Wrel  = (const float*)d_in[4 + 3 * l];
        const float* Wroot = (const float*)d_in[5 + 3 * l];
        const float* bias  = (const float*)d_in[6 + 3 * l];

        // zero aggregation buffer
        zero_f32<<<cdiv((long long)N * fin, THREADS), THREADS, 0, stream>>>(agg, N * fin);
        // scatter-add weighted neighbor features
        edge_scatter<<<cdiv((long long)E * fin, THREADS), THREADS, 0, stream>>>(
            hin, ei + src_off, ei + dst_off, ea, agg, E, fin);
        // fused dual-GEMM + bias + relu via f32 WMMA
        const int tiles = (N / 16) * (fout / 16);
        gconv_gemm<<<cdiv(tiles, THREADS / 32), THREADS, 0, stream>>>(
            agg, hin, Wrel, Wroot, bias, hout, N, fin, fout);

        hin = hout;
        hout = (hout == hA) ? hB : hA;
    }

    // global mean pool (C = 256)
    const int C = 256;
    zero_f32<<<cdiv(G * C + G, THREADS), THREADS, 0, stream>>>(gbuf0, G * C);
    zero_f32<<<1, THREADS, 0, stream>>>(counts, G);
    pool_sum<<<cdiv((long long)N * C, THREADS), THREADS, 0, stream>>>(hin, bat, gbuf0, counts, N, C);
    pool_div<<<cdiv(G * C, THREADS), THREADS, 0, stream>>>(gbuf0, counts, G, C);

    // MLP head: 256 -> 256 -> 128 -> 64 -> 1
    static const int MD[5] = {256, 256, 128, 64, 1};
    const float* min_ = gbuf0;
    float* mbufs[2] = {gbuf1, gbuf0};
    for (int l = 0; l < 4; ++l) {
        const float* W = (const float*)d_in[25 + 2 * l];
        const float* b = (const float*)d_in[26 + 2 * l];
        float* mout = (l == 3) ? (float*)d_out : mbufs[l & 1];
        mlp_layer<<<cdiv(G * MD[l + 1], THREADS), THREADS, 0, stream>>>(
            min_, W, b, mout, G, MD[l], MD[l + 1], l < 3 ? 1 : 0);
        min_ = mout;
    }
}